// GeodesicDistance_10746008175281
// MI455X (gfx1250) — compile-verified
//
#include <hip/hip_runtime.h>
#include <cmath>

typedef float v2f __attribute__((ext_vector_type(2)));
typedef float v8f __attribute__((ext_vector_type(8)));

// ---------------------------------------------------------------------------
// Cross-lane helpers (wave32). SWAPX16: xor=0x10, or=0, and=0x1f -> 0x401f.
// ---------------------------------------------------------------------------
__device__ __forceinline__ float swap16(float x) {
  return __int_as_float(__builtin_amdgcn_ds_swizzle(__float_as_int(x), 0x401f));
}

__device__ __forceinline__ v2f splat2(float x) {
  v2f r; r.x = x; r.y = x; return r;
}

// Canonical row vector: 8 register pairs; pair c IS the WMMA B-operand for
// K-chunk c (lane L<16 holds j=4c,4c+1; lane L+16 holds j=4c+2,4c+3 of row
// L%16). Keeping it in v2f makes every elementwise op a v_pk_fma_f32 and
// every WMMA B operand a pre-aligned register pair (no marshalling movs).
struct RVec {
  v2f c[8];
};

// Row-wise dot product, result replicated into both lane halves.
__device__ __forceinline__ float rowdot(const RVec& a, const RVec& b) {
  v2f acc = {0.f, 0.f};
#pragma unroll
  for (int i = 0; i < 8; ++i) acc += a.c[i] * b.c[i];
  float s = acc.x + acc.y;
  return s + swap16(s);
}

// Convert WMMA D-layout (two 16x16 f32 tiles: component i = 16R + 8h + d at
// lane-half h, acc reg d) into canonical layout. Component i stays in-lane
// iff bit1(i)==bit3(i); otherwise it crosses via SWAPX16.
__device__ __forceinline__ void d2canon(bool lo, const v8f& D0, const v8f& D1,
                                        RVec& out) {
#pragma unroll
  for (int t = 0; t < 2; ++t) {
    const v8f& D = t ? D1 : D0;
#pragma unroll
    for (int kk = 0; kk < 4; ++kk) {
      const int d0 = (kk < 2) ? kk : kk + 2;  // 0,1,4,5
      float sw_a = swap16(D[d0]);             // executed by all lanes
      float sw_b = swap16(D[d0 + 2]);
      float vlo = lo ? D[d0] : sw_b;          // canonical k = 8t+kk
      float vhi = lo ? sw_a : D[d0 + 2];      // canonical k = 8t+4+kk
      const int pl = 4 * t + (kk >> 1);       // pair of k=8t+kk
      const int ph = pl + 2;                  // pair of k=8t+4+kk
      if (kk & 1) { out.c[pl].y = vlo; out.c[ph].y = vhi; }
      else        { out.c[pl].x = vlo; out.c[ph].x = vhi; }
    }
  }
}

// w = A*q for 16 rows: 16 x v_wmma_f32_16x16x4_f32 accumulating two 16x16
// D tiles (output components 0..15 / 16..31), then layout conversion.
__device__ __forceinline__ void matvecA(const v2f aop[2][8], const RVec& in,
                                        RVec& out, bool lo) {
  v8f acc0 = {0.f, 0.f, 0.f, 0.f, 0.f, 0.f, 0.f, 0.f};
  v8f acc1 = {0.f, 0.f, 0.f, 0.f, 0.f, 0.f, 0.f, 0.f};
#pragma unroll
  for (int c = 0; c < 8; ++c) {
    acc0 = __builtin_amdgcn_wmma_f32_16x16x4_f32(false, aop[0][c], false,
                                                 in.c[c], (short)0, acc0,
                                                 false, false);
    acc1 = __builtin_amdgcn_wmma_f32_16x16x4_f32(false, aop[1][c], false,
                                                 in.c[c], (short)0, acc1,
                                                 false, false);
  }
  d2canon(lo, acc0, acc1, out);
}

struct GeoParams {
  float dt[5];
  int   np[5];
  float inv1[5];  // 1/(1 - 0.9^t)
  float inv2[5];  // 1/(1 - 0.999^t)
};

// ---------------------------------------------------------------------------
// Prologue kernels
// ---------------------------------------------------------------------------
__global__ void geo_init(unsigned* ws) {
  if (threadIdx.x == 0) ws[0] = 0x7f800000u;  // +inf bit pattern
}

// A = W * W^T  (32x32, row-major)
__global__ void geo_gram(const float* __restrict__ W, float* __restrict__ A) {
  for (int idx = threadIdx.x; idx < 1024; idx += blockDim.x) {
    int i = idx >> 5, j = idx & 31;
    float s = 0.f;
#pragma unroll
    for (int k = 0; k < 32; ++k) s = fmaf(W[i * 32 + k], W[j * 32 + k], s);
    A[idx] = s;
  }
}

// Global min of lam(q) = 1 + ||qW||^2 over q0 and q1. lam >= 1 > 0, so the
// uint bit pattern preserves float ordering -> atomicMin on bits.
__global__ __launch_bounds__(256) void geo_lammin(
    const float* __restrict__ q0, const float* __restrict__ q1,
    const float* __restrict__ W, unsigned* __restrict__ minout) {
  __shared__ v2f Ws2[512];  // Ws2[k*16+jp] = {W[k][2jp], W[k][2jp+1]}
  __shared__ float red[256];
  for (int i = threadIdx.x; i < 512; i += blockDim.x)
    Ws2[i] = *(const v2f*)(W + 2 * i);
  __syncthreads();
  int row = blockIdx.x * blockDim.x + threadIdx.x;
  float lmin = 3.4e38f;
  const float* arr[2] = {q0, q1};
#pragma unroll
  for (int a = 0; a < 2; ++a) {
    const float* qr = arr[a] + (size_t)row * 32;
    float qv[32];
#pragma unroll
    for (int k = 0; k < 32; ++k) qv[k] = qr[k];
    v2f acc = {0.f, 0.f};
#pragma unroll 2
    for (int jp = 0; jp < 16; ++jp) {
      v2f u = {0.f, 0.f};
#pragma unroll
      for (int k = 0; k < 32; ++k) u += splat2(qv[k]) * Ws2[k * 16 + jp];
      acc += u * u;
    }
    lmin = fminf(lmin, 1.0f + acc.x + acc.y);
  }
  red[threadIdx.x] = lmin;
  __syncthreads();
  for (int s = 128; s > 0; s >>= 1) {
    if (threadIdx.x < s)
      red[threadIdx.x] = fminf(red[threadIdx.x], red[threadIdx.x + s]);
    __syncthreads();
  }
  if (threadIdx.x == 0) atomicMin(minout, __float_as_uint(red[0]));
}

// ---------------------------------------------------------------------------
// Main kernel: one wave (32 threads) per 16 batch rows.
// Euler step: w=Aq; lam=1+q.w; pn=p.p; p'=p-2dt*pn*w; q'=q+2dt*lam*p
// Adjoint:    pb = pb - 4dt(pb.w)p + 2dt*lam*qb
//             qb = qb - 2dt*pn*(A pb) + 4dt(qb.p)w
// ---------------------------------------------------------------------------
__global__ __launch_bounds__(32) void geo_main(
    const float* __restrict__ q0g, const float* __restrict__ q1g,
    const float* __restrict__ Ag, const unsigned* __restrict__ minbits,
    float* __restrict__ outg, GeoParams P) {
  // Checkpoint stack: 8 steps x {p(8 pairs), q(8 pairs)} x 32 lanes = 32 KB.
  __shared__ v2f lds2[4096];
  const int lane = threadIdx.x & 31;
  const int r = lane & 15;
  const int hh = lane >> 4;
  const bool lo = (hh == 0);
  const size_t row = (size_t)blockIdx.x * 16 + r;

  // A operands (A row-major 32x32): lane holds A[16R+r][4c+2hh + {0,1}].
  v2f aop[2][8];
#pragma unroll
  for (int R = 0; R < 2; ++R)
#pragma unroll
    for (int c = 0; c < 8; ++c)
      aop[R][c] = *(const v2f*)(Ag + (R * 16 + r) * 32 + 4 * c + 2 * hh);

  // Load q0, q1 straight into canonical layout (8-byte loads).
  RVec q0v, q1v, p0;
#pragma unroll
  for (int c = 0; c < 8; ++c) {
    q0v.c[c] = *(const v2f*)(q0g + row * 32 + 4 * c + 2 * hh);
    q1v.c[c] = *(const v2f*)(q1g + row * 32 + 4 * c + 2 * hh);
    p0.c[c] = splat2(0.5f) * (q1v.c[c] - q0v.c[c]);
  }

  // new_lr from global min lam:  mf = gmin^-16
  float gmin = __uint_as_float(minbits[0]);
  float ti = 1.0f / gmin;
  float t2 = ti * ti, t4 = t2 * t2, t8 = t4 * t4;
  float mf = t8 * t8;
  float new_lr = 0.1f * ((1.0f - expf(-0.5f * mf)) * 5.0f + 0.01f);

  const float gscale = 2.0f / (131072.0f * 32.0f);  // d(mean sq)/dq seed

  RVec m, v;
#pragma unroll
  for (int c = 0; c < 8; ++c) {
    m.c[c] = splat2(0.f);
    v.c[c] = splat2(0.f);
  }

#pragma unroll
  for (int it = 0; it < 5; ++it) {
    const float dt = P.dt[it];
    const int n = P.np[it];

    RVec p = p0, q = q0v;

    // ---- forward shoot, checkpoint (p,q) per step (ds_store_b64) ----
    for (int s = 0; s < n; ++s) {
#pragma unroll
      for (int c = 0; c < 8; ++c) {
        lds2[(s * 16 + c) * 32 + lane] = p.c[c];
        lds2[(s * 16 + 8 + c) * 32 + lane] = q.c[c];
      }
      RVec w;
      matvecA(aop, q, w, lo);
      float lam = 1.0f + rowdot(q, w);
      float pn = rowdot(p, p);
      v2f cq = splat2(2.0f * dt * lam);
      v2f cp = splat2(2.0f * dt * pn);
#pragma unroll
      for (int c = 0; c < 8; ++c) {
        v2f po = p.c[c];
        p.c[c] = po - cp * w.c[c];
        q.c[c] = q.c[c] + cq * po;
      }
    }

    // ---- adjoint seed at the loss ----
    RVec pb, qb;
    v2f gs = splat2(gscale);
#pragma unroll
    for (int c = 0; c < 8; ++c) {
      qb.c[c] = gs * (q.c[c] - q1v.c[c]);
      pb.c[c] = splat2(0.f);
    }

    // ---- backward through the trajectory ----
    for (int s = n - 1; s >= 0; --s) {
      RVec ps, qs;
#pragma unroll
      for (int c = 0; c < 8; ++c) {
        ps.c[c] = lds2[(s * 16 + c) * 32 + lane];
        qs.c[c] = lds2[(s * 16 + 8 + c) * 32 + lane];
      }
      RVec w, tt;
      matvecA(aop, qs, w, lo);   // recompute w = A q_s
      matvecA(aop, pb, tt, lo);  // A * pb'
      float lam = 1.0f + rowdot(qs, w);
      float pn = rowdot(ps, ps);
      float d1 = rowdot(pb, w);
      float d2 = rowdot(qb, ps);
      v2f c1 = splat2(4.0f * dt * d1), c2 = splat2(2.0f * dt * lam);
      v2f c3 = splat2(2.0f * dt * pn), c4 = splat2(4.0f * dt * d2);
#pragma unroll
      for (int c = 0; c < 8; ++c) {
        v2f pbn = pb.c[c] - c1 * ps.c[c] + c2 * qb.c[c];
        v2f qbn = qb.c[c] - c3 * tt.c[c] + c4 * w.c[c];
        pb.c[c] = pbn;
        qb.c[c] = qbn;
      }
    }

    // ---- Adam update of p0 (g = pb) ----
    const float i1 = P.inv1[it], i2 = P.inv2[it];
#pragma unroll
    for (int c = 0; c < 8; ++c) {
      v2f g = pb.c[c];
      m.c[c] = splat2(0.9f) * m.c[c] + splat2(0.1f) * g;
      v.c[c] = splat2(0.999f) * v.c[c] + splat2(0.001f) * (g * g);
#pragma unroll
      for (int e = 0; e < 2; ++e) {
        float mh = m.c[c][e] * i1;
        float vh = v.c[c][e] * i2;
        p0.c[c][e] = p0.c[c][e] - new_lr * mh / (sqrtf(vh) + 1e-8f);
      }
    }
  }

  // ---- H0 = lam(q0) * ||p0||^2 ; out = sqrt(max(H0,0)) ----
  RVec w;
  matvecA(aop, q0v, w, lo);
  float lam0 = 1.0f + rowdot(q0v, w);
  float pn0 = rowdot(p0, p0);
  float H = lam0 * pn0;
  if (lo) outg[row] = sqrtf(fmaxf(H, 0.0f));
}

// ---------------------------------------------------------------------------
extern "C" void kernel_launch(void* const* d_in, const int* in_sizes, int n_in,
                              void* d_out, int out_size, void* d_ws,
                              size_t ws_size, hipStream_t stream) {
  (void)n_in; (void)out_size; (void)ws_size;
  const float* q0 = (const float*)d_in[0];
  const float* q1 = (const float*)d_in[1];
  const float* W = (const float*)d_in[2];
  float* out = (float*)d_out;
  unsigned* minbits = (unsigned*)d_ws;  // ws[0]: lam-min bits
  float* A = (float*)d_ws + 16;         // ws[16..16+1024): A = W W^T

  const int B = in_sizes[0] / 32;  // 131072

  // Host-side schedule (double cos with 3.1415 to match math.cos exactly).
  GeoParams P;
  double b1p = 1.0, b2p = 1.0;
  for (int it = 0; it < 5; ++it) {
    double x = (double)it / 5.0;
    double sched = 1.0 + (5.0 - 1.0) * (1.0 + cos(x * 3.1415)) / 2.0;
    double dtd = 0.1 * sched;
    P.dt[it] = (float)dtd;
    P.np[it] = (int)ceil(1.0 / dtd);  // {2,3,3,5,8}
    b1p *= 0.9;
    b2p *= 0.999;
    P.inv1[it] = (float)(1.0 / (1.0 - b1p));
    P.inv2[it] = (float)(1.0 / (1.0 - b2p));
  }

  geo_init<<<1, 32, 0, stream>>>(minbits);
  geo_gram<<<1, 256, 0, stream>>>(W, A);
  geo_lammin<<<B / 256, 256, 0, stream>>>(q0, q1, W, minbits);
  geo_main<<<B / 16, 32, 0, stream>>>(q0, q1, A, minbits, out, P);
}